// MixtureOfExperts_27900107555055
// MI455X (gfx1250) — compile-verified
//
#include <hip/hip_runtime.h>
#include <hip/hip_bf16.h>

// Problem constants (from reference): B=4, S=512, D=128, A=64, E=8, TOP_K=2
#define BB     4
#define SS     512
#define DD     128
#define AA     64
#define EE     8
#define NTOK   (BB * SS)

typedef float v2f __attribute__((ext_vector_type(2)));
typedef float v8f __attribute__((ext_vector_type(8)));

// ---------------------------------------------------------------------------
// Kernel 1: gating (per batch), top-2 selection, balance loss, adaptive bias
// One tiny block; all work is O(KB).
// ---------------------------------------------------------------------------
__global__ __launch_bounds__(128) void moe_prepare(
    const float* __restrict__ adapt,    // [B][A]
    const float* __restrict__ gate_W,   // [E][A]
    const float* __restrict__ gate_b,   // [E]
    const float* __restrict__ ab_W,     // [E][D][A]
    const float* __restrict__ ab_b,     // [E][D]
    int*         __restrict__ sel,      // [B][2]  (workspace)
    float*       __restrict__ cbias,    // [B][D]  (workspace)
    float*       __restrict__ loss_out) // &d_out[N*D]
{
    __shared__ float sc[BB][EE];
    __shared__ int   ssel[BB][2];
    const int t = threadIdx.x;

    // gate scores: 32 (b,e) dot products of length A
    if (t < BB * EE) {
        const int b = t >> 3, e = t & 7;
        const float* a = adapt  + b * AA;
        const float* w = gate_W + e * AA;
        float s = gate_b[e];
        #pragma unroll 8
        for (int i = 0; i < AA; ++i) s += a[i] * w[i];
        sc[b][e] = s;
    }
    __syncthreads();

    // top-2 per batch (softmax is monotone -> argmax of raw scores;
    // first-index-wins on ties, matching lax.top_k)
    if (t < BB) {
        const int b = t;
        int i0 = 0; float v0 = sc[b][0];
        for (int e = 1; e < EE; ++e) if (sc[b][e] > v0) { v0 = sc[b][e]; i0 = e; }
        int i1 = (i0 == 0) ? 1 : 0; float v1 = sc[b][i1];
        for (int e = 0; e < EE; ++e)
            if (e != i0 && sc[b][e] > v1) { v1 = sc[b][e]; i1 = e; }
        ssel[b][0] = i0; ssel[b][1] = i1;
        sel[b * 2 + 0] = i0; sel[b * 2 + 1] = i1;
    }
    __syncthreads();

    if (t == 0) {
        // usage[e] = S * (#batches selecting e); expected = N*K/E = 512
        float loss = 0.f;
        for (int e = 0; e < EE; ++e) {
            int cnt = 0;
            for (int b = 0; b < BB; ++b)
                cnt += (ssel[b][0] == e) + (ssel[b][1] == e);
            const float d = (float)SS * (float)cnt - (float)(NTOK * 2 / EE);
            loss += d * d;
        }
        loss_out[0] = loss;
    }

    // adaptive bias, summed over ALL experts (reference adds it unmasked)
    const int o = t;  // blockDim == D
    for (int b = 0; b < BB; ++b) {
        const float* a = adapt + b * AA;
        float acc = 0.f;
        for (int e = 0; e < EE; ++e) {
            const float* w = ab_W + (e * DD + o) * AA;
            float s = ab_b[e * DD + o];
            #pragma unroll 8
            for (int i = 0; i < AA; ++i) s += a[i] * w[i];
            acc += s;
        }
        cbias[b * DD + o] = acc;
    }
}

// ---------------------------------------------------------------------------
// Kernel 2: build per-batch effective matrix in a WMMA-B-fragment-friendly
// K-pair-interleaved layout:
//   Mtp[b][d>>1][o][d&1] = sum_{j<2, e=sel[b][j]} ( weight[e][o][d]
//                          + aw_b[e][o*D+d] + dot(aw_W[e][o*D+d][:], ad_b) )
// so a lane's B fragment (M[o][k], M[o][k+1]) is ONE contiguous b64 load.
// Bandwidth-bound GEMV over the selected aw_W slices (the dominant traffic).
// ---------------------------------------------------------------------------
__global__ __launch_bounds__(256) void moe_build_M(
    const float* __restrict__ adapt,   // [B][A]
    const float* __restrict__ weight,  // [E][D][D]
    const float* __restrict__ aw_W,    // [E][D*D][A]
    const float* __restrict__ aw_b,    // [E][D*D]
    const int*   __restrict__ sel,     // [B][2]
    float*       __restrict__ Mtp)     // [B][D/2][D][2]
{
    __shared__ float ads[AA];
    const int b   = blockIdx.y;
    const int idx = blockIdx.x * 256 + threadIdx.x;   // 0 .. D*D-1
    if (threadIdx.x < AA) ads[threadIdx.x] = adapt[b * AA + threadIdx.x];
    __syncthreads();

    const int o = idx >> 7;     // idx / D
    const int d = idx & (DD - 1);

    float val = 0.f;
    #pragma unroll
    for (int j = 0; j < 2; ++j) {
        const int e = sel[b * 2 + j];
        val += weight[((size_t)e * DD + o) * DD + d] + aw_b[(size_t)e * DD * DD + idx];
        const float4* w = (const float4*)(aw_W + ((size_t)e * DD * DD + idx) * AA);
        float s = 0.f;
        #pragma unroll
        for (int i = 0; i < AA / 4; ++i) {
            const float4 wv = w[i];
            s += wv.x * ads[i * 4 + 0] + wv.y * ads[i * 4 + 1]
               + wv.z * ads[i * 4 + 2] + wv.w * ads[i * 4 + 3];
        }
        val += s;
    }
    // K-pair interleaved store (scatter, but stores are 256 KB vs 32 MB reads)
    Mtp[(((size_t)b * (DD / 2) + (d >> 1)) * DD + o) * 2 + (d & 1)] = val;
}

// ---------------------------------------------------------------------------
// Kernel 3: per-token GEMM with WMMA fp32:
//   out[b, m, o] = sum_d X[b, m, d] * M_b[o, d] + cbias[b, o]
// Block = 128 threads (4 waves). Block owns a 16-row tile; each wave owns two
// 16-col tiles. K=128 chained over 32 x v_wmma_f32_16x16x4_f32, C seeded with
// the bias. A fragments from LDS (ds_load_2addr_b64, conflict-free via +4
// padding); B fragments are single global_load_b64 thanks to the pair layout.
// ---------------------------------------------------------------------------
__global__ __launch_bounds__(128) void moe_gemm_wmma(
    const float* __restrict__ x,      // [B][S][D]
    const float* __restrict__ Mtp,    // [B][D/2][D][2]
    const float* __restrict__ cbias,  // [B][D]
    float*       __restrict__ out)    // [B][S][D]
{
    __shared__ float Xs[16][DD + 4];  // +4 words: conflict-free column reads

    const int b    = blockIdx.y;
    const int row0 = blockIdx.x * 16;
    const int t    = threadIdx.x;
    const float* Xb = x + ((size_t)b * SS + row0) * DD;

    // Cooperative, fully-coalesced load of the 16x128 X tile (512 float4).
    #pragma unroll
    for (int i = 0; i < 4; ++i) {
        const int fidx = t + i * 128;        // float4 index
        const int r    = fidx >> 5;          // 32 float4 per row
        const int c4   = fidx & 31;
        const float4 v = ((const float4*)(Xb + r * DD))[c4];
        Xs[r][c4 * 4 + 0] = v.x; Xs[r][c4 * 4 + 1] = v.y;
        Xs[r][c4 * 4 + 2] = v.z; Xs[r][c4 * 4 + 3] = v.w;
    }
    __syncthreads();

    const int wave = t >> 5;
    const int lane = t & 31;
    const int half = lane >> 4;   // K half-select per ISA 16x4 A layout
    const int l16  = lane & 15;
    const float2* Mp = (const float2*)Mtp + (size_t)b * (DD / 2) * DD;

    #pragma unroll
    for (int cti = 0; cti < 2; ++cti) {
        const int col0 = (wave + cti * 4) * 16;

        // Seed C with the bias: C element (M=r+8*half, N=l16) -> bias[col0+l16]
        const float cb = cbias[b * DD + col0 + l16];
        v8f acc;
        #pragma unroll
        for (int r = 0; r < 8; ++r) acc[r] = cb;

        #pragma unroll
        for (int k0 = 0; k0 < DD; k0 += 4) {
            const int kk = k0 + 2 * half;          // this half's K base
            v2f a, bf;
            // A 16x4 layout: lanes 0-15 rows M, VGPR0/1 = K, K+1; halves split K by 2
            a.x = Xs[l16][kk];
            a.y = Xs[l16][kk + 1];
            // B fragment: (M[o][kk], M[o][kk+1]) is one contiguous float2
            const float2 bp = Mp[(size_t)((k0 >> 1) + half) * DD + col0 + l16];
            bf.x = bp.x;
            bf.y = bp.y;
            acc = __builtin_amdgcn_wmma_f32_16x16x4_f32(
                /*neg_a=*/false, a, /*neg_b=*/false, bf,
                /*c_mod=*/(short)0, acc, /*reuse_a=*/false, /*reuse_b=*/false);
        }

        float* outb = out + ((size_t)b * SS + row0) * DD + col0;
        #pragma unroll
        for (int r = 0; r < 8; ++r) {
            const int m = r + 8 * half;      // C layout: lanes 16-31 hold M=8..15
            outb[(size_t)m * DD + l16] = acc[r];
        }
    }
}

// ---------------------------------------------------------------------------
extern "C" void kernel_launch(void* const* d_in, const int* in_sizes, int n_in,
                              void* d_out, int out_size, void* d_ws, size_t ws_size,
                              hipStream_t stream) {
    (void)in_sizes; (void)n_in; (void)out_size; (void)ws_size;

    const float* x      = (const float*)d_in[0];  // [B,S,D]
    const float* adapt  = (const float*)d_in[1];  // [B,A]
    const float* weight = (const float*)d_in[2];  // [E,D,D]
    const float* aw_W   = (const float*)d_in[3];  // [E,D*D,A]
    const float* aw_b   = (const float*)d_in[4];  // [E,D*D]
    const float* ab_W   = (const float*)d_in[5];  // [E,D,A]
    const float* ab_b   = (const float*)d_in[6];  // [E,D]
    const float* gate_W = (const float*)d_in[7];  // [E,A]
    const float* gate_b = (const float*)d_in[8];  // [E]

    float* out  = (float*)d_out;                  // [B,S,D] then [1] loss
    float* loss = out + (size_t)NTOK * DD;

    // workspace layout (all offsets 8-byte aligned)
    int*   sel   = (int*)d_ws;                    // 8 ints
    float* cbias = (float*)d_ws + 64;             // [B][D]
    float* Mtp   = (float*)d_ws + 64 + BB * DD;   // [B][D/2][D][2]

    moe_prepare<<<1, 128, 0, stream>>>(adapt, gate_W, gate_b, ab_W, ab_b,
                                       sel, cbias, loss);
    moe_build_M<<<dim3((DD * DD) / 256, BB), 256, 0, stream>>>(
        adapt, weight, aw_W, aw_b, sel, Mtp);
    moe_gemm_wmma<<<dim3(SS / 16, BB), 128, 0, stream>>>(x, Mtp, cbias, out);
}